// NormalFloatQuantizer_82798379532989
// MI455X (gfx1250) — compile-verified
//
#include <hip/hip_runtime.h>
#include <stdint.h>

#define NF4_BS       512
#define WG_WAVES     8
#define WG_THREADS   (WG_WAVES * 32)   // 256 threads = 8 wave32
#define TILES_PER_WG 4                 // tiles (8 NF4 blocks each) per workgroup

typedef float f32x4 __attribute__((ext_vector_type(4)));
typedef int   i32x4v __attribute__((vector_size(16)));   // matches builtin pointee type

// ---------------- CDNA5 async global->LDS path (probe-guarded) ----------------
#if defined(__HIP_DEVICE_COMPILE__) && __has_builtin(__builtin_amdgcn_global_load_async_to_lds_b128)
#define USE_ASYNC 1
#else
#define USE_ASYNC 0
#endif

#if USE_ASYNC
#if __has_builtin(__builtin_amdgcn_s_wait_asynccnt)
#define WAIT_ASYNC(n) __builtin_amdgcn_s_wait_asynccnt(n)
#else
#define WAIT_ASYNC(n) asm volatile("s_wait_asynccnt " #n ::: "memory")
#endif
#else
#define WAIT_ASYNC(n) do {} while (0)
#endif

#define LDS_FENCE() asm volatile("" ::: "memory")

__device__ __forceinline__ void async_copy16(const float* g, float* l) {
#if USE_ASYNC
    __builtin_amdgcn_global_load_async_to_lds_b128(
        (__attribute__((address_space(1))) i32x4v*)(g),
        (__attribute__((address_space(3))) i32x4v*)(l),
        0, 0);
#else
    *(f32x4*)l = *(const f32x4*)g;
#endif
}

// ---------------- NF4 codebook ----------------
__device__ __constant__ float kNF4[16] = {
    -1.0f, -0.6961928009986877f, -0.5250730514526367f, -0.39491748809814453f,
    -0.28444138169288635f, -0.18477343022823334f, -0.09105003625154495f, 0.0f,
    0.07958029955625534f, 0.16093020141124725f, 0.24611230194568634f,
    0.33791524171829224f, 0.4407098591327667f, 0.5626170039176941f,
    0.7229568362236023f, 1.0f};

// idx = #{ mid_i < xn }  ==  jnp.searchsorted(MID, xn, side='left')
__device__ __forceinline__ int nf4_index(float xn) {
    int i = 0;
    i += xn > -0.8480964004993439f;
    i += xn > -0.6106329262256622f;
    i += xn > -0.4599952697753906f;
    i += xn > -0.3396794348955154f;
    i += xn > -0.2346074059605598f;
    i += xn > -0.1379117332398891f;
    i += xn > -0.0455250181257725f;
    i += xn >  0.0397901497781277f;
    i += xn >  0.1202552504837513f;
    i += xn >  0.2035212516784668f;
    i += xn >  0.2920137718319893f;
    i += xn >  0.3893125504255295f;
    i += xn >  0.5016634315252304f;
    i += xn >  0.6427869200706482f;
    i += xn >  0.8614784181118011f;
    return i;
}

__device__ __forceinline__ void issue_tile(const float* __restrict__ x, float* buf,
                                           int tile, int wave, int lane, int nblocks) {
    int blk = tile * WG_WAVES + wave;
    blk = (blk < nblocks) ? blk : (nblocks - 1);   // clamp: keeps ASYNCcnt bookkeeping exact
    const float* gsrc = x + (size_t)blk * NF4_BS + lane * 4;
    float* ldst = buf + lane * 4;
#pragma unroll
    for (int c = 0; c < 4; ++c)
        async_copy16(gsrc + c * 128, ldst + c * 128);
}

__device__ __forceinline__ void compute_block(const float* buf, float* __restrict__ out,
                                              const float* lut,
                                              int tile, int wave, int lane, int nblocks) {
    const int blk = tile * WG_WAVES + wave;
    f32x4 v[4];
#pragma unroll
    for (int c = 0; c < 4; ++c)
        v[c] = *(const f32x4*)(buf + lane * 4 + c * 128);   // ds_load_b128

    // per-lane absmax over 16 values
    float m = 0.0f;
#pragma unroll
    for (int c = 0; c < 4; ++c) {
        m = fmaxf(m, fabsf(v[c].x));
        m = fmaxf(m, fabsf(v[c].y));
        m = fmaxf(m, fabsf(v[c].z));
        m = fmaxf(m, fabsf(v[c].w));
    }
    // wave32 butterfly reduction
#pragma unroll
    for (int off = 16; off > 0; off >>= 1)
        m = fmaxf(m, __shfl_xor(m, off, 32));

    const float scale = (m == 0.0f) ? 1.0f : m;
    const float inv = 1.0f / scale;   // one divide per lane, amortized over 16 elems

    f32x4 r[4];
#pragma unroll
    for (int c = 0; c < 4; ++c) {
        r[c].x = lut[nf4_index(v[c].x * inv)] * m;
        r[c].y = lut[nf4_index(v[c].y * inv)] * m;
        r[c].z = lut[nf4_index(v[c].z * inv)] * m;
        r[c].w = lut[nf4_index(v[c].w * inv)] * m;
    }

    if (blk < nblocks) {
        float* dst = out + (size_t)blk * NF4_BS + lane * 4;
#pragma unroll
        for (int c = 0; c < 4; ++c)
            __builtin_nontemporal_store(r[c], (f32x4*)(dst + c * 128));
    }
}

__global__ void __launch_bounds__(WG_THREADS)
nf4_quant_dequant_kernel(const float* __restrict__ x, float* __restrict__ out,
                         int nblocks, int ntiles) {
    __shared__ float lut[16];
    __shared__ __align__(16) float stage[2][WG_WAVES][NF4_BS];   // 32 KB double buffer

    const int tid  = threadIdx.x;
    const int wave = tid >> 5;
    const int lane = tid & 31;

    if (tid < 16) lut[tid] = kNF4[tid];

    const int tile0 = blockIdx.x * TILES_PER_WG;
    if (tile0 < ntiles)
        issue_tile(x, stage[0][wave], tile0, wave, lane, nblocks);   // prologue

    __syncthreads();   // LUT visible to all waves

    for (int t = 0; t < TILES_PER_WG; ++t) {
        const int tile = tile0 + t;
        if (tile >= ntiles) break;
        const bool have_next = (t + 1 < TILES_PER_WG) && (tile + 1 < ntiles);
        if (have_next)
            issue_tile(x, stage[(t + 1) & 1][wave], tile + 1, wave, lane, nblocks);
        // async loads complete in order: <=4 outstanding means current tile's 4 are done
        if (have_next) { WAIT_ASYNC(4); } else { WAIT_ASYNC(0); }
        LDS_FENCE();
        compute_block(stage[t & 1][wave], out, lut, tile, wave, lane, nblocks);
    }
}

extern "C" void kernel_launch(void* const* d_in, const int* in_sizes, int n_in,
                              void* d_out, int out_size, void* d_ws, size_t ws_size,
                              hipStream_t stream) {
    const float* x = (const float*)d_in[0];
    float* out = (float*)d_out;
    const int n = in_sizes[0];
    const int nblocks = n / NF4_BS;
    const int ntiles = (nblocks + WG_WAVES - 1) / WG_WAVES;
    const int grid = (ntiles + TILES_PER_WG - 1) / TILES_PER_WG;
    nf4_quant_dequant_kernel<<<grid, WG_THREADS, 0, stream>>>(x, out, nblocks, ntiles);
}